// LocalityLoss_39462159515958
// MI455X (gfx1250) — compile-verified
//
#include <hip/hip_runtime.h>
#include <hip/hip_bf16.h>
#include <math.h>

#define EPS 1e-6f
#define T_DIM 32
#define D_DIM 256
#define H_DIM 56
#define W_DIM 56
#define CD 8                      // d-slices per block in the reduction pass
#define CHUNKS (D_DIM / CD)       // 32

typedef __attribute__((ext_vector_type(2))) float v2f;
typedef __attribute__((ext_vector_type(8))) float v8f;

// workspace partial layout: part[a][t][chunk][bin], a: 0=s1h 1=s2h 2=s1w 3=s2w
#define PART_IDX(a, t, c, b) ((((a)*T_DIM + (t))*CHUNKS + (c))*H_DIM + (b))

// ---------------------------------------------------------------------------
// Kernel 1: streaming reduction. One block = (t, 8 d-slices). Bandwidth-bound:
// coalesced b32 row loads, register accumulation for w-bins, wave shuffle
// reduction for h-bins. No atomics anywhere -> bitwise deterministic.
// ---------------------------------------------------------------------------
__global__ __launch_bounds__(256) void locality_reduce(const float* __restrict__ f,
                                                       float* __restrict__ part) {
  __shared__ float sh_h1[H_DIM];
  __shared__ float sh_h2[H_DIM];
  __shared__ float w1p[8][W_DIM];
  __shared__ float w2p[8][W_DIM];

  const int t     = blockIdx.x / CHUNKS;
  const int chunk = blockIdx.x % CHUNKS;
  const int d0    = chunk * CD;
  const int tid   = threadIdx.x;
  const int wv    = tid >> 5;
  const int lane  = tid & 31;

  if (tid < H_DIM)               sh_h1[tid] = 0.f;
  else if (tid < 2 * H_DIM)      sh_h2[tid - H_DIM] = 0.f;
  __syncthreads();

  const float* base = f + ((size_t)t * D_DIM + d0) * (size_t)(H_DIM * W_DIM);

  float cw1a = 0.f, cw2a = 0.f;   // w = lane
  float cw1b = 0.f, cw2b = 0.f;   // w = 32 + lane (lanes 0..23)

  // wave wv owns rows h == wv (mod 8); 56 rows total over the 8 slices
  int h = wv, d = 0;
  for (int it = 0; it < (CD * H_DIM) / 8; ++it) {
    const float* row = base + (size_t)(d * H_DIM + h) * W_DIM;
    float v0 = row[lane];
    float v1 = (lane < (W_DIM - 32)) ? row[32 + lane] : 0.f;

    cw1a += v0;  cw2a = fmaf(v0, v0, cw2a);
    cw1b += v1;  cw2b = fmaf(v1, v1, cw2b);

    float rs1 = v0 + v1;
    float rs2 = fmaf(v0, v0, v1 * v1);
#pragma unroll
    for (int off = 16; off > 0; off >>= 1) {
      rs1 += __shfl_xor(rs1, off);
      rs2 += __shfl_xor(rs2, off);
    }
    if (lane == 0) {              // h-bin owned exclusively by this wave
      sh_h1[h] += rs1;
      sh_h2[h] += rs2;
    }
    h += 8;
    if (h >= H_DIM) { h -= H_DIM; ++d; }
  }

  w1p[wv][lane] = cw1a;
  w2p[wv][lane] = cw2a;
  if (lane < (W_DIM - 32)) {
    w1p[wv][32 + lane] = cw1b;
    w2p[wv][32 + lane] = cw2b;
  }
  __syncthreads();

  if (tid < 4 * H_DIM) {          // 224 partial values per block
    int a = tid / H_DIM, b = tid % H_DIM;
    float v;
    if (a == 0)      v = sh_h1[b];
    else if (a == 1) v = sh_h2[b];
    else {
      v = 0.f;
      if (a == 2) { for (int k = 0; k < 8; ++k) v += w1p[k][b]; }
      else        { for (int k = 0; k < 8; ++k) v += w2p[k][b]; }
    }
    part[PART_IDX(a, t, chunk, b)] = v;
  }
}

// ---------------------------------------------------------------------------
// Kernel 2: finalize. Fixed-order partial accumulation, then batched
// prefix/suffix cumsums as WMMA f32 matmuls with triangular ones matrices:
//   pre[t,j] = sum_{i<=j} S[t,i]  =  (S * U)[t,j],  U[i,j] = (i<=j)
//   suf[t,j] = sum_{i>=j} S[t,i]  =  (S * L)[t,j],  L[i,j] = (i>=j)
// S is [32 x 64] (56 bins zero-padded). Epilogue: sqrt / log10 / mean.
// ---------------------------------------------------------------------------
__global__ __launch_bounds__(256) void locality_finalize(const float* __restrict__ part,
                                                         float* __restrict__ out) {
  __shared__ float S[4][T_DIM][66];   // stride 66 to dodge bank conflicts
  __shared__ float red[256];

  const int tid = threadIdx.x;

  // Gather chunk partials (fixed order => deterministic), zero-pad K to 64.
  for (int idx = tid; idx < 4 * T_DIM * 64; idx += 256) {
    int a   = idx >> 11;            // / (32*64)
    int rem = idx & 2047;
    int t   = rem >> 6;
    int j   = rem & 63;
    float v = 0.f;
    if (j < H_DIM) {
      const float* p = part + PART_IDX(a, t, 0, j);
#pragma unroll
      for (int c = 0; c < CHUNKS; ++c) v += p[c * H_DIM];
    }
    S[a][t][j] = v;
  }
  __syncthreads();

  const int wv   = tid >> 5;
  const int lane = tid & 31;
  const int g    = wv & 3;              // 0..3 -> ga1..ga4
  const int tr   = wv >> 2;             // t-tile: rows 0-15 or 16-31
  const int a1   = (g < 2) ? 0 : 2;     // s1 array (h or w)
  const int a2   = a1 + 1;              // matching s2 array
  const bool suffix = (g == 0) || (g == 2);

  const int   m    = lane & 15;         // A row within tile / B-D column
  const int   half = lane >> 4;
  const float nper = (float)(D_DIM * W_DIM);   // D*W == D*H == 14336

  float local = 0.f;

  for (int tc = 0; tc < 4; ++tc) {      // column tiles over j = 0..63
    v8f acc1 = {};                      // S1 * M
    v8f acc2 = {};                      // S2 * M
    const int j = tc * 16 + m;          // global output bin for this lane

#pragma unroll
    for (int kk = 0; kk < 16; ++kk) {   // K = 64 in steps of 4
      const int k0 = kk * 4 + 2 * half; // ISA A-layout: lanes 0-15 K={0,1}, 16-31 K={2,3}
      v2f A1 = { S[a1][tr * 16 + m][k0], S[a1][tr * 16 + m][k0 + 1] };
      v2f A2 = { S[a2][tr * 16 + m][k0], S[a2][tr * 16 + m][k0 + 1] };
      float b0, b1;
      if (suffix) { b0 = (k0     >= j) ? 1.f : 0.f; b1 = (k0 + 1 >= j) ? 1.f : 0.f; }
      else        { b0 = (k0     <= j) ? 1.f : 0.f; b1 = (k0 + 1 <= j) ? 1.f : 0.f; }
      v2f B = { b0, b1 };
      acc1 = __builtin_amdgcn_wmma_f32_16x16x4_f32(false, A1, false, B, (short)0,
                                                   acc1, false, false);
      acc2 = __builtin_amdgcn_wmma_f32_16x16x4_f32(false, A2, false, B, (short)0,
                                                   acc2, false, false);
    }

    if (j < H_DIM) {
      // pairwise_distance(x,0,p=2,eps): sqrt(sum x^2 + 2*eps*sum x + eps^2*n)
      const float ncnt = suffix ? nper * (float)(H_DIM - j)
                                : nper * (float)(j + 1);
#pragma unroll
      for (int e = 0; e < 8; ++e) {     // D-layout: VGPR e -> row tr*16 + half*8 + e
        const int trow = tr * 16 + half * 8 + e;
        const float val = sqrtf(acc2[e] + 2.f * EPS * acc1[e] + EPS * EPS * ncnt);
        out[1 + g * (T_DIM * H_DIM) + trow * H_DIM + j] = log10f(val);
        local += val + EPS;             // contribution to tot
      }
    }
  }

  red[tid] = local;
  __syncthreads();
  if (tid == 0) {
    float s = 0.f;
    for (int i = 0; i < 256; ++i) s += red[i];   // fixed order -> deterministic
    out[0] = s / (4.f * (float)T_DIM);
  }
}

// ---------------------------------------------------------------------------
extern "C" void kernel_launch(void* const* d_in, const int* in_sizes, int n_in,
                              void* d_out, int out_size, void* d_ws, size_t ws_size,
                              hipStream_t stream) {
  const float* f   = (const float*)d_in[0];
  float*       out = (float*)d_out;
  float*       part = (float*)d_ws;   // 4*32*32*56 floats = 896 KiB

  locality_reduce<<<T_DIM * CHUNKS, 256, 0, stream>>>(f, part);
  locality_finalize<<<1, 256, 0, stream>>>(part, out);
}